// MultivariateMuyGPs_layer_78726750536170
// MI455X (gfx1250) — compile-verified
//
#include <hip/hip_runtime.h>

// Problem constants (from reference)
#define TRAIN   60000
#define FDIM    16
#define BATCH   2000
#define NNB     64
#define MMOD    10
#define SQRT3F  1.7320508075688772f

typedef __attribute__((ext_vector_type(2))) float v2f;
typedef __attribute__((ext_vector_type(8))) float v8f;

__device__ __forceinline__ float matern15(float d, float inv_ls) {
    float z = SQRT3F * d * inv_ls;
    return (1.0f + z) * __expf(-z);
}

// One workgroup (256 threads = 8 wave32) per batch element b.
// Phase 1: stage features, Gram via V_WMMA_F32_16X16X4_F32, pairwise distances.
// Phase 2: 5 sweeps x 2 concurrent models; per model: build Matern kernel,
//          LDL-style Gaussian elimination with fused RHS elimination
//          (no triangular solves needed: results are pivot-weighted dots).
__global__ __launch_bounds__(256)
void muygps_main_kernel(const float* __restrict__ x,            // [TRAIN,F]
                        const float* __restrict__ length_scale, // [M]
                        const float* __restrict__ eps,          // [M]
                        const float* __restrict__ targets,      // [B,NN,M]
                        const int*   __restrict__ bidx,         // [B]
                        const int*   __restrict__ nnidx,        // [B,NN]
                        float*       __restrict__ pred_out,     // [B,M]
                        float*       __restrict__ quad_ws,      // [B,M]
                        float*       __restrict__ ydot_ws)      // [B,M]
{
    __shared__ float Xs[NNB][FDIM + 1];     // neighbor features (padded)
    __shared__ float dist[NNB][NNB];        // Gram -> pairwise distance
    __shared__ float Kp[2][NNB][NNB + 1];   // two models' kernel matrices (pitch 65)
    __shared__ float w1[2][NNB];            // eliminated RHS: targets
    __shared__ float w2[2][NNB];            // eliminated RHS: Kcross
    __shared__ float dcross[NNB];
    __shared__ float nrm[NNB];

    const int b    = blockIdx.x;
    const int tid  = threadIdx.x;
    const int lane = tid & 31;
    const int wave = tid >> 5;

    // ---- Stage neighbor features into LDS (64 rows x 16 cols) ----
    for (int i = tid; i < NNB * FDIM; i += 256) {
        int r = i >> 4, c = i & 15;
        Xs[r][c] = x[nnidx[b * NNB + r] * FDIM + c];
    }
    __syncthreads();

    // ---- Crosswise distances: query vs each neighbor ----
    if (tid < NNB) {
        const float* xb = x + bidx[b] * FDIM;
        float s = 0.0f;
        #pragma unroll
        for (int c = 0; c < FDIM; ++c) {
            float d = xb[c] - Xs[tid][c];
            s += d * d;
        }
        dcross[tid] = sqrtf(s);
    }

    // ---- Gram matrix G = X * X^T via V_WMMA_F32_16X16X4_F32 ----
    // 4x4 grid of 16x16 tiles, K=16 in four K=4 steps. 2 tiles per wave,
    // fully unrolled: straight-line code, EXEC all ones as WMMA requires.
    {
        const int row   = lane & 15;          // fragment row (M for A, N for B)
        const int khalf = (lane >> 4) * 2;    // lanes 0-15: K=0,1; lanes 16-31: K=2,3
        const int n     = lane & 15;
        const int mbase = (lane >> 4) * 8;
        #pragma unroll
        for (int tt = 0; tt < 2; ++tt) {
            const int t  = wave + tt * 8;     // tile id 0..15
            const int ti = t >> 2, tj = t & 3;
            v8f acc = {};
            #pragma unroll
            for (int k = 0; k < FDIM; k += 4) {
                v2f a, bb;
                a.x  = Xs[ti * 16 + row][k + khalf + 0];
                a.y  = Xs[ti * 16 + row][k + khalf + 1];
                bb.x = Xs[tj * 16 + row][k + khalf + 0];
                bb.y = Xs[tj * 16 + row][k + khalf + 1];
                acc = __builtin_amdgcn_wmma_f32_16x16x4_f32(
                    false, a, false, bb, (short)0, acc, false, false);
            }
            #pragma unroll
            for (int v = 0; v < 8; ++v) {
                dist[ti * 16 + mbase + v][tj * 16 + n] = acc[v];
            }
        }
    }
    __syncthreads();

    // ---- Gram -> pairwise L2 distance ----
    if (tid < NNB) nrm[tid] = dist[tid][tid];
    __syncthreads();
    for (int i = tid; i < NNB * NNB; i += 256) {
        int r = i >> 6, c = i & 63;
        float s = nrm[r] + nrm[c] - 2.0f * dist[r][c];
        dist[r][c] = sqrtf(fmaxf(s, 0.0f));
    }
    __syncthreads();

    // ---- 5 sweeps x 2 models, LDL elimination with fused RHS ----
    const int mg = tid >> 7;       // model slot within sweep: 0 or 1
    const int g  = tid & 127;      // local thread id within model group
    const int i  = g & 63;         // owned row
    const int colbase = (g >> 6) * 32;  // owned half-row: cols [colbase, colbase+32)

    for (int sweep = 0; sweep < 5; ++sweep) {
        const int mo = sweep * 2 + mg;             // model index 0..9
        const float il = 1.0f / length_scale[mo];
        const float ep = eps[mo];

        // Build Kp = matern(dist) + eps*I  (full symmetric storage)
        for (int idx = g; idx < NNB * NNB; idx += 128) {
            int r = idx >> 6, c = idx & 63;
            float k = matern15(dist[r][c], il);
            if (r == c) k += ep;
            Kp[mg][r][c] = k;
        }
        if (g < NNB) {
            w1[mg][g] = targets[(b * NNB + g) * MMOD + mo];
            w2[mg][g] = matern15(dcross[g], il);
        }
        __syncthreads();

        // Gaussian elimination (SPD, no pivoting) with fused RHS elimination.
        // One barrier per step; pivots remain on the diagonal.
        for (int k = 0; k < NNB; ++k) {
            const bool active = (i > k);
            if (active) {
                const float invp = __builtin_amdgcn_rcpf(Kp[mg][k][k]);
                const float mik  = Kp[mg][i][k] * invp;   // multiplier L[i][k]
                if (colbase == 0) {
                    w1[mg][i] -= mik * w1[mg][k];
                    w2[mg][i] -= mik * w2[mg][k];
                }
                const int j0 = (k + 1 > colbase) ? (k + 1) : colbase;
                const int j1 = colbase + 32;
                for (int j = j0; j < j1; ++j) {
                    Kp[mg][i][j] -= mik * Kp[mg][k][j];   // row-k broadcast read
                }
            }
            __syncthreads();
        }

        // pred = sum w1*w2/p, quad = sum w2^2/p, ydot = sum w1^2/p
        if (g == 0) {
            float sp = 0.0f, sq = 0.0f, sy = 0.0f;
            for (int k = 0; k < NNB; ++k) {
                float ip = 1.0f / Kp[mg][k][k];
                float a  = w1[mg][k];
                float c  = w2[mg][k];
                sp += a * c * ip;
                sq += c * c * ip;
                sy += a * a * ip;
            }
            pred_out[b * MMOD + mo] = sp;
            quad_ws [b * MMOD + mo] = sq;
            ydot_ws [b * MMOD + mo] = sy;
        }
        __syncthreads();   // protect Kp/w before next sweep overwrites
    }
}

// Deterministic per-model reduction of ydot over batches -> sigma^2[m]
__global__ __launch_bounds__(256)
void muygps_sigma_kernel(const float* __restrict__ ydot_ws,  // [B,M]
                         float* __restrict__ sigma_ws)       // [M]
{
    __shared__ float red[256];
    const int m = blockIdx.x;
    const int tid = threadIdx.x;
    float s = 0.0f;
    for (int b = tid; b < BATCH; b += 256) s += ydot_ws[b * MMOD + m];
    red[tid] = s;
    __syncthreads();
    for (int st = 128; st > 0; st >>= 1) {
        if (tid < st) red[tid] += red[tid + st];
        __syncthreads();
    }
    if (tid == 0) sigma_ws[m] = red[0] / (float)(BATCH * NNB);
}

// variances[b,m] = (1 - quad[b,m]) * sigma^2[m]
__global__ __launch_bounds__(256)
void muygps_var_kernel(const float* __restrict__ quad_ws,
                       const float* __restrict__ sigma_ws,
                       float* __restrict__ var_out)
{
    int i = blockIdx.x * blockDim.x + threadIdx.x;
    if (i < BATCH * MMOD) {
        int m = i % MMOD;
        var_out[i] = (1.0f - quad_ws[i]) * sigma_ws[m];
    }
}

extern "C" void kernel_launch(void* const* d_in, const int* in_sizes, int n_in,
                              void* d_out, int out_size, void* d_ws, size_t ws_size,
                              hipStream_t stream) {
    const float* x            = (const float*)d_in[0];
    const float* length_scale = (const float*)d_in[1];
    const float* eps          = (const float*)d_in[2];
    const float* targets      = (const float*)d_in[3];
    const int*   bidx         = (const int*)d_in[4];
    const int*   nnidx        = (const int*)d_in[5];

    float* pred_out = (float*)d_out;                    // [B,M]
    float* var_out  = (float*)d_out + BATCH * MMOD;     // [B,M]

    float* quad_ws  = (float*)d_ws;                     // [B,M]
    float* ydot_ws  = quad_ws + BATCH * MMOD;           // [B,M]
    float* sigma_ws = ydot_ws + BATCH * MMOD;           // [M]

    muygps_main_kernel<<<BATCH, 256, 0, stream>>>(
        x, length_scale, eps, targets, bidx, nnidx, pred_out, quad_ws, ydot_ws);

    muygps_sigma_kernel<<<MMOD, 256, 0, stream>>>(ydot_ws, sigma_ws);

    muygps_var_kernel<<<(BATCH * MMOD + 255) / 256, 256, 0, stream>>>(
        quad_ws, sigma_ws, var_out);
}